// Attention_6399501271287
// MI455X (gfx1250) — compile-verified
//
#include <hip/hip_runtime.h>
#include <hip/hip_bf16.h>

typedef __attribute__((ext_vector_type(2))) float v2f;
typedef __attribute__((ext_vector_type(8))) float v8f;
typedef __attribute__((ext_vector_type(4))) unsigned int v4u;
typedef __attribute__((ext_vector_type(4))) int v4i;
typedef __attribute__((ext_vector_type(8))) int v8i;

#define D_FEAT 64
#define ROWSTRIDE 68   // 64 + 4 pad floats: bank = (4*row + col) % 64 -> conflict-free
#define WAVES 2        // waves per block (64 threads)

#if defined(__has_builtin)
#if __has_builtin(__builtin_amdgcn_tensor_load_to_lds)
#define HAVE_TDM 1
#endif
#endif

// ---------------------------------------------------------------------------
// Kernel 1: h_norm[n] = sum_d (Y[n,d] * etas[d])^2   (half-wave per node)
// ---------------------------------------------------------------------------
__global__ __launch_bounds__(256) void hnorm_kernel(const float* __restrict__ Y,
                                                    const float* __restrict__ etas,
                                                    float* __restrict__ hn, int N) {
    int gid  = blockIdx.x * blockDim.x + threadIdx.x;
    int node = gid >> 4;
    int q    = gid & 15;
    if (node >= N) return;
    float4 e4 = ((const float4*)etas)[q];
    float4 y4 = ((const float4*)(Y + (size_t)node * D_FEAT))[q];
    float a = y4.x * e4.x, b = y4.y * e4.y, c = y4.z * e4.z, d = y4.w * e4.w;
    float p = a * a + b * b + c * c + d * d;
    p += __shfl_xor(p, 8, 16);
    p += __shfl_xor(p, 4, 16);
    p += __shfl_xor(p, 2, 16);
    p += __shfl_xor(p, 1, 16);
    if (q == 0) hn[node] = p;
}

// ---------------------------------------------------------------------------
// Kernel 2: zero deg region of d_out (harness poisons it with 0xAA)
// ---------------------------------------------------------------------------
__global__ __launch_bounds__(256) void zero_kernel(float* __restrict__ p, int n) {
    int i = blockIdx.x * blockDim.x + threadIdx.x;
    if (i < n) p[i] = 0.0f;
}

#if HAVE_TDM
// ---------------------------------------------------------------------------
// TDM gather-mode descriptor: pull 16 indexed 64-float rows of Y into LDS,
// padding LDS by 4 DWORDs every 64 DWORDs (=> ROWSTRIDE 68-float rows).
// D# layout per CDNA5 ISA 8.3-8.7 (gather mode, 16-bit row indices).
// This toolchain's builtin is the 6-arg form:
//   (uint32x4 g0, int32x8 g1, int32x4 g2, int32x4 g3, int32x8 g4, i32 cpol)
// g4 corresponds to the ISA's unused VADDR4 slot -> pass zeros.
// ---------------------------------------------------------------------------
__device__ __forceinline__ void tdm_gather16(unsigned ldsoff, unsigned long long gbase,
                                             const unsigned* pk, int N) {
    v4u g0;
    g0.x = 0x80000001u;                                   // count=1, gather_mode=1, idx16
    g0.y = ldsoff;                                        // lds_addr (bytes)
    g0.z = (unsigned)gbase;                               // global_addr[31:0]
    g0.w = (unsigned)((gbase >> 32) & 0x01FFFFFFull)      // global_addr[56:32]
         | 0x80000000u;                                   // type=2 ("image")
    v8i g1;
    g1[0] = (int)0x07520000u;   // data_size=4B, pad_enable, pad_interval=64dw, pad_amount=4dw
    g1[1] = (int)(64u << 16);                             // tensor_dim0 = 64
    g1[2] = (int)(((unsigned)N & 0xFFFFu) << 16);         // tensor_dim1[15:0] = N lo
    g1[3] = (int)((((unsigned)N >> 16) & 0xFFFFu)         // tensor_dim1[31:16]
         | (64u << 16));                                  // tile_dim0 = 64
    g1[4] = 16;                                           // tile_dim1 = #row indices
    g1[5] = 64;                                           // tensor_dim0_stride = 64
    g1[6] = 0;
    g1[7] = 0;
    v4i g2, g3;                                           // 16 x 16-bit row indices
    g2[0] = (int)pk[0]; g2[1] = (int)pk[1]; g2[2] = (int)pk[2]; g2[3] = (int)pk[3];
    g3[0] = (int)pk[4]; g3[1] = (int)pk[5]; g3[2] = (int)pk[6]; g3[3] = (int)pk[7];
    v8i g4 = {0, 0, 0, 0, 0, 0, 0, 0};                    // unused VADDR4 group
    __builtin_amdgcn_tensor_load_to_lds(g0, g1, g2, g3, g4, 0);
}
#endif

// ---------------------------------------------------------------------------
// Kernel 3: per-edge weight via WMMA f32 16x16x4 (diag of (U.eta2) * V^T),
// TDM gather staging (double-buffered), epilogue + atomic scatter into deg.
// Block = 64 threads = 2 waves; each wave owns a 16-edge tile.
// ---------------------------------------------------------------------------
__global__ __launch_bounds__(64) void edge_wmma_kernel(const float* __restrict__ Y,
                                                       const float* __restrict__ etas,
                                                       const int* __restrict__ src,
                                                       const int* __restrict__ dst,
                                                       const float* __restrict__ hn,
                                                       float* __restrict__ wOut,
                                                       float* __restrict__ deg,
                                                       int N, int E, int tiles) {
    // rows 0..15 = src rows, 16..31 = dst rows (RAW Y values; eta^2 applied in A frag)
    __shared__ __align__(16) float rows[WAVES][2][32][ROWSTRIDE];
    __shared__ float eta2[D_FEAT];
#if !HAVE_TDM
    __shared__ int nidx[WAVES][32];
#endif

    const int tid  = threadIdx.x;
    const int wave = __builtin_amdgcn_readfirstlane(tid >> 5);
    const int l    = tid & 31;
    const int half = l >> 4;
    const int q    = l & 15;

    {   float e = etas[tid & 63];  eta2[tid & 63] = e * e; }
    __syncthreads();

#if HAVE_TDM
    const unsigned long long gbase = (unsigned long long)(size_t)Y;
    // Issue both 16-row gathers (src-side, dst-side) for tile t into buffer buf.
    auto issue_tile = [&](int t, int buf) {
        const int base = t * (WAVES * 16) + wave * 16;
        unsigned pkS[8], pkD[8];
        #pragma unroll
        for (int i = 0; i < 8; ++i) {
            const int e0 = base + 2 * i, e1 = e0 + 1;
            unsigned s0 = (e0 < E) ? (unsigned)src[e0] : 0u;
            unsigned s1 = (e1 < E) ? (unsigned)src[e1] : 0u;
            unsigned d0 = (e0 < E) ? (unsigned)dst[e0] : 0u;
            unsigned d1 = (e1 < E) ? (unsigned)dst[e1] : 0u;
            pkS[i] = (unsigned)__builtin_amdgcn_readfirstlane(
                         (int)((s0 & 0xFFFFu) | (s1 << 16)));
            pkD[i] = (unsigned)__builtin_amdgcn_readfirstlane(
                         (int)((d0 & 0xFFFFu) | (d1 << 16)));
        }
        const unsigned off = (unsigned)(size_t)&rows[wave][buf][0][0];
        tdm_gather16(off, gbase, pkS, N);
        tdm_gather16(off + 16u * ROWSTRIDE * 4u, gbase, pkD, N);
    };

    const int t0 = blockIdx.x;
    if (t0 < tiles) issue_tile(t0, 0);
    int it = 0;
    for (int t = t0; t < tiles; t += gridDim.x, ++it) {
        const int buf = it & 1;
        const int tn  = t + gridDim.x;
        if (tn < tiles) {
            issue_tile(tn, buf ^ 1);                     // prefetch next tile
            __builtin_amdgcn_s_wait_tensorcnt(2);        // current tile's pair done
        } else {
            __builtin_amdgcn_s_wait_tensorcnt(0);
        }
        asm volatile("" ::: "memory");
#else
    for (int t = blockIdx.x; t < tiles; t += gridDim.x) {
        const int buf = 0;
        {   // fallback staging: global -> VGPR -> LDS
            const int base = t * (WAVES * 16) + wave * 16;
            const int e = base + q;
            int sv_ = 0, dv_ = 0;
            if (e < E) { sv_ = src[e]; dv_ = dst[e]; }
            if (half == 0) { nidx[wave][q] = sv_; nidx[wave][16 + q] = dv_; }
            __syncthreads();
            #pragma unroll
            for (int s = 0; s < 16; ++s) {
                const int rr   = 2 * s + half;
                const int node = nidx[wave][rr];
                float4 g = ((const float4*)(Y + (size_t)node * D_FEAT))[q];
                ((float4*)&rows[wave][buf][rr][4 * q])[0] = g;
            }
            __syncthreads();
        }
#endif
        const int base = t * (WAVES * 16) + wave * 16;
        const int e    = base + q;

        // D = (U .* eta^2) * V^T via 16 chained V_WMMA_F32_16X16X4_F32 (K=64)
        v8f c = {};
        #pragma unroll
        for (int s2 = 0; s2 < 16; ++s2) {
            const int kb = 4 * s2 + 2 * half;
            const float* ua = &rows[wave][buf][q][kb];       // src row q (raw)
            const float* vb = &rows[wave][buf][16 + q][kb];  // dst row q (raw)
            const float* ep = &eta2[kb];
            v2f a, b;
            a.x = ua[0] * ep[0]; a.y = ua[1] * ep[1];
            b.x = vb[0];         b.y = vb[1];
            c = __builtin_amdgcn_wmma_f32_16x16x4_f32(false, a, false, b,
                                                      (short)0, c, false, false);
        }

        // Diagonal extract: D[m][m] in lane m (vgpr m) for m<8, lane m+16 (vgpr m-8) else
        const int sel = l & 7;
        float v = c[0];
        v = (sel == 1) ? c[1] : v;
        v = (sel == 2) ? c[2] : v;
        v = (sel == 3) ? c[3] : v;
        v = (sel == 4) ? c[4] : v;
        v = (sel == 5) ? c[5] : v;
        v = (sel == 6) ? c[6] : v;
        v = (sel == 7) ? c[7] : v;
        const int srcLane = (q < 8) ? q : (q + 16);
        const float dotv = __shfl(v, srcLane, 32);

        if (half == 0 && e < E) {
            const int su = src[e];
            const int du = dst[e];
            float w = hn[su] + hn[du] - 2.0f * dotv;   // ||y_u - y_v||^2
            w = fmaxf(w, 0.0f) + 1e-7f;                 // relu + eps
            w = sqrtf(w);                               // w^(1 - 0.5*P), P=1
            w = fmaxf(w, 0.1f);                         // TAU
            w = (w > 5.0f) ? 0.0f : (1.0f / w);         // T_UP threshold + reciprocal
            w += 1e-9f;
            wOut[e] = w;
            atomicAdd(&deg[du], w);                     // global_atomic_add_f32
        }
#if !HAVE_TDM
        __syncthreads();
#endif
    }
}

// ---------------------------------------------------------------------------
extern "C" void kernel_launch(void* const* d_in, const int* in_sizes, int n_in,
                              void* d_out, int out_size, void* d_ws, size_t ws_size,
                              hipStream_t stream) {
    const float* Y    = (const float*)d_in[0];
    const float* etas = (const float*)d_in[1];
    const int*   src  = (const int*)d_in[2];
    const int*   dst  = (const int*)d_in[3];

    const int N = in_sizes[0] / D_FEAT;   // 50000
    const int E = in_sizes[2];            // 800000

    float* wOut = (float*)d_out;          // [E]
    float* deg  = wOut + E;               // [N]
    float* hn   = (float*)d_ws;           // [N] scratch

    {   // h_norm
        int threads = N * 16;
        int blocks  = (threads + 255) / 256;
        hnorm_kernel<<<blocks, 256, 0, stream>>>(Y, etas, hn, N);
    }
    {   // zero deg accumulator
        int blocks = (N + 255) / 256;
        zero_kernel<<<blocks, 256, 0, stream>>>(deg, N);
    }
    {   // edge weights + scatter (32 edges per block tile)
        int tiles  = (E + 31) / 32;
        int blocks = tiles < 4096 ? tiles : 4096;
        edge_wmma_kernel<<<blocks, 64, 0, stream>>>(Y, etas, src, dst, hn,
                                                    wOut, deg, N, E, tiles);
    }
}